// MidspaceTransform_2087354106268
// MI455X (gfx1250) — compile-verified
//
#include <hip/hip_runtime.h>
#include <stdint.h>

// ---------------------------------------------------------------------------
// Batched principal sqrt of 1M near-identity 4x4 affine matrices (top 3x4 out).
//
// Structure: A = [[L, t],[0,1]] (block upper triangular, spectrum near 1).
// Principal sqrt: sqrt(A) = [[S, u],[0,1]] with S = sqrt(L), (S+I) u = t.
// So: run inverse-free Newton-Schulz (Higham eq. 6.35) on the 3x3 block only
//   Y0 = L, Z0 = I;  T = 0.5*(3I - Z*Y);  Y <- Y*T;  Z <- T*Z;  Y -> sqrt(L)
// Error recurrence delta' ~ (3/4) delta^2 with worst delta0 ~ 0.3 over 1M
// samples: 4 full steps reach fp32 roundoff; we run 6 effective iterations
// (first specialized for Z0=I, last with Z dead) for ~1.5 steps of margin,
// then recover u with one 3x3 adjugate solve of (S+I) u = t
// (S+I ~ 2I: condition ~1, det ~ 8, division is safe).
//
// Mapping: one matrix per lane (wave32), everything in registers; VOPD packs
// the FMA stream. Kernel is VALU-bound (memory floor ~4.1us at 23.3 TB/s),
// so the front end uses the gfx1250 async path: global_load_async_to_lds_b128
// into a per-wave LDS double buffer + s_wait_asynccnt, overlapping the next
// tile's 1536B fetch with the current tile's compute.
// WMMA intentionally unused: batched 3x3/4x4 with a sequential dependence
// chain maps to fp32 WMMA 16x16x4 at 25% utilization plus cross-lane
// repacking per product (C layout != A/B layouts) -- 2-3x worse than the
// per-lane FMA path.
// ---------------------------------------------------------------------------

#define NS_ITERS 6                      /* total effective NS iterations */
#define WAVES_PER_BLOCK 8
#define BLOCK_THREADS (WAVES_PER_BLOCK * 32)
#define LDS_STRIDE_F 20                 /* floats per lane slot: 80B, 16B-aligned */
#define BUF_F (32 * LDS_STRIDE_F)       /* floats per wave buffer */

struct M33 { float m[3][3]; };

__device__ __forceinline__ M33 mul33(const M33& A, const M33& B) {
  M33 C;
#pragma unroll
  for (int i = 0; i < 3; ++i) {
#pragma unroll
    for (int j = 0; j < 3; ++j) {
      float s = A.m[i][0] * B.m[0][j];
      s = fmaf(A.m[i][1], B.m[1][j], s);
      s = fmaf(A.m[i][2], B.m[2][j], s);
      C.m[i][j] = s;
    }
  }
  return C;
}

// T = 1.5*I - 0.5*M
__device__ __forceinline__ M33 half_3I_minus(const M33& M) {
  M33 T;
#pragma unroll
  for (int i = 0; i < 3; ++i)
#pragma unroll
    for (int j = 0; j < 3; ++j)
      T.m[i][j] = fmaf(-0.5f, M.m[i][j], (i == j) ? 1.5f : 0.0f);
  return T;
}

__global__ __launch_bounds__(BLOCK_THREADS)
void midspace_sqrtm_kernel(const float* __restrict__ src,
                           float* __restrict__ dst, int nmat) {
  __shared__ __align__(16) float smem[WAVES_PER_BLOCK * 2 * BUF_F];

  const unsigned lane   = threadIdx.x & 31u;
  const unsigned wib    = threadIdx.x >> 5;
  const unsigned waves  = (gridDim.x * blockDim.x) >> 5;
  const unsigned gwave  = (blockIdx.x * blockDim.x + threadIdx.x) >> 5;
  const unsigned ntiles = ((unsigned)nmat + 31u) >> 5;

  float* laneBuf0 = &smem[(wib * 2u + 0u) * BUF_F + lane * LDS_STRIDE_F];
  float* laneBuf1 = &smem[(wib * 2u + 1u) * BUF_F + lane * LDS_STRIDE_F];
  // Generic LDS address carries the LDS byte offset in bits [31:0] (aperture).
  const unsigned ldsOff0 = (unsigned)(uintptr_t)(const void*)laneBuf0;
  const unsigned ldsOff1 = (unsigned)(uintptr_t)(const void*)laneBuf1;

  // One tile = 32 matrices = 48B/lane, fetched as 3 async b128 copies to LDS.
  // offset: applies to both the global and the LDS address (ISA async pseudocode).
  auto issue_tile = [&](unsigned t, unsigned lOff) {
    unsigned m = t * 32u + lane;
    if (m >= (unsigned)nmat) m = (unsigned)nmat - 1u;   // tail clamp (no OOB)
    const unsigned gOff = m * 48u;                      // 48MB < 2^31, fits u32
    asm volatile(
        "global_load_async_to_lds_b128 %0, %1, %2 offset:0\n\t"
        "global_load_async_to_lds_b128 %0, %1, %2 offset:16\n\t"
        "global_load_async_to_lds_b128 %0, %1, %2 offset:32"
        :
        : "v"(lOff), "v"(gOff), "s"(src)
        : "memory");
  };

  unsigned tile = gwave;
  if (tile < ntiles) issue_tile(tile, ldsOff0);
  unsigned cur = 0;

  while (tile < ntiles) {
    const unsigned nxt = tile + waves;
    const bool has_next = nxt < ntiles;
    if (has_next) {
      issue_tile(nxt, cur ? ldsOff0 : ldsOff1);
      // 3 ops outstanding for the next tile; async loads retire in order, so
      // cnt<=3 guarantees the current tile's 3 copies have landed in LDS.
      asm volatile("s_wait_asynccnt 0x3" ::: "memory");
    } else {
      asm volatile("s_wait_asynccnt 0x0" ::: "memory");
    }

    const float4* lp = reinterpret_cast<const float4*>(cur ? laneBuf1 : laneBuf0);
    const float4 r0 = lp[0];          // L[0][0..2], t0
    const float4 r1 = lp[1];          // L[1][0..2], t1
    const float4 r2 = lp[2];          // L[2][0..2], t2
    const float t0 = r0.w, t1 = r1.w, t2 = r2.w;

    M33 Y;
    Y.m[0][0]=r0.x; Y.m[0][1]=r0.y; Y.m[0][2]=r0.z;
    Y.m[1][0]=r1.x; Y.m[1][1]=r1.y; Y.m[1][2]=r1.z;
    Y.m[2][0]=r2.x; Y.m[2][1]=r2.y; Y.m[2][2]=r2.z;

    // --- iteration 1 specialized (Z0 = I  =>  M = Y, Z1 = T) ---------------
    M33 Z = half_3I_minus(Y);         // T
    Y = mul33(Y, Z);                  // Y1 = L*T

    // --- iterations 2 .. NS_ITERS-1 (full coupled update) ------------------
#pragma unroll
    for (int it = 0; it < NS_ITERS - 2; ++it) {
      M33 P = mul33(Z, Y);            // Z*Y
      M33 T = half_3I_minus(P);
      M33 Yn = mul33(Y, T);           // Y <- Y*T
      M33 Zn = mul33(T, Z);           // Z <- T*Z
      Y = Yn; Z = Zn;
    }

    // --- last iteration: Z dead afterwards, update Y only ------------------
    {
      M33 P = mul33(Z, Y);
      M33 T = half_3I_minus(P);
      Y = mul33(Y, T);                // S = sqrt(L)
    }

    // --- translation: u = (S + I)^{-1} t via adjugate (B ~ 2I, det ~ 8) ----
    const float b00 = Y.m[0][0] + 1.0f, b01 = Y.m[0][1],        b02 = Y.m[0][2];
    const float b10 = Y.m[1][0],        b11 = Y.m[1][1] + 1.0f, b12 = Y.m[1][2];
    const float b20 = Y.m[2][0],        b21 = Y.m[2][1],        b22 = Y.m[2][2] + 1.0f;

    const float a00 = fmaf(b11, b22, -b12 * b21);
    const float a01 = fmaf(b02, b21, -b01 * b22);
    const float a02 = fmaf(b01, b12, -b02 * b11);
    const float a10 = fmaf(b12, b20, -b10 * b22);
    const float a11 = fmaf(b00, b22, -b02 * b20);
    const float a12 = fmaf(b02, b10, -b00 * b12);
    const float a20 = fmaf(b10, b21, -b11 * b20);
    const float a21 = fmaf(b01, b20, -b00 * b21);
    const float a22 = fmaf(b00, b11, -b01 * b10);

    const float det  = fmaf(b00, a00, fmaf(b01, a10, b02 * a20));
    const float rdet = 1.0f / det;

    const float u0 = fmaf(a00, t0, fmaf(a01, t1, a02 * t2)) * rdet;
    const float u1 = fmaf(a10, t0, fmaf(a11, t1, a12 * t2)) * rdet;
    const float u2 = fmaf(a20, t0, fmaf(a21, t1, a22 * t2)) * rdet;

    const unsigned midx = tile * 32u + lane;
    if (midx < (unsigned)nmat) {
      float4* op = reinterpret_cast<float4*>(dst + (size_t)midx * 12u);
      op[0] = make_float4(Y.m[0][0], Y.m[0][1], Y.m[0][2], u0);
      op[1] = make_float4(Y.m[1][0], Y.m[1][1], Y.m[1][2], u1);
      op[2] = make_float4(Y.m[2][0], Y.m[2][1], Y.m[2][2], u2);
    }

    tile = nxt;
    cur ^= 1u;
  }
}

extern "C" void kernel_launch(void* const* d_in, const int* in_sizes, int n_in,
                              void* d_out, int out_size, void* d_ws, size_t ws_size,
                              hipStream_t stream) {
  (void)n_in; (void)out_size; (void)d_ws; (void)ws_size;
  const float* src = (const float*)d_in[0];
  float* dst = (float*)d_out;
  const int nmat = in_sizes[0] / 12;                      // [B,3,4] fp32
  const unsigned ntiles = ((unsigned)nmat + 31u) >> 5;
  unsigned blocks = (ntiles + WAVES_PER_BLOCK - 1u) / WAVES_PER_BLOCK;
  if (blocks > 1024u) blocks = 1024u;   // 8192 waves; each pipelines ~4 tiles
  if (blocks == 0u) blocks = 1u;
  midspace_sqrtm_kernel<<<dim3(blocks), dim3(BLOCK_THREADS), 0, stream>>>(src, dst, nmat);
}